// GlobalRGCNEncoder_91104846283009
// MI455X (gfx1250) — compile-verified
//
#include <hip/hip_runtime.h>
#include <hip/hip_bf16.h>

#define DIM 128
#define ID_DIM 112
#define ROLE_DIM 16
#define NREL 3
#define KTOT 512   // 3*128 (relations) + 128 (root)

typedef __attribute__((ext_vector_type(2))) float v2f;
typedef __attribute__((ext_vector_type(8))) float v8f;

// Guaranteed hardware FP32 atomic add (avoids any CAS-loop fallback).
__device__ __forceinline__ void atomic_fadd(float* p, float v) {
  asm volatile("global_atomic_add_f32 %0, %1, off" : : "v"(p), "v"(v) : "memory");
}

// x[n, 0:112] = id_emb[n], x[n, 112:128] = role_emb[roles[n]]
__global__ __launch_bounds__(256) void build_x_kernel(
    const float* __restrict__ id_emb, const float* __restrict__ role_emb,
    const int* __restrict__ roles, float* __restrict__ x, int n) {
  int gid = blockIdx.x * blockDim.x + threadIdx.x;
  if (gid >= n * DIM) return;
  int node = gid >> 7;
  int d = gid & (DIM - 1);
  float v;
  if (d < ID_DIM) v = id_emb[(size_t)node * ID_DIM + d];
  else            v = role_emb[roles[node] * ROLE_DIM + (d - ID_DIM)];
  x[gid] = v;
}

// cnt[et*N + dst] += 1 per edge (edges identical for both layers; done once)
__global__ __launch_bounds__(256) void count_kernel(
    const int* __restrict__ dst, const int* __restrict__ et,
    int* __restrict__ cnt, int E, int n) {
  int e = blockIdx.x * blockDim.x + threadIdx.x;
  if (e >= E) return;
  atomicAdd(&cnt[et[e] * n + dst[e]], 1);
}

// in-place int count -> float reciprocal scale 1/max(cnt,1)
__global__ __launch_bounds__(256) void scale_kernel(float* __restrict__ sc, int m) {
  int i = blockIdx.x * blockDim.x + threadIdx.x;
  if (i >= m) return;
  int c = ((const int*)sc)[i];
  sc[i] = 1.0f / (float)(c > 0 ? c : 1);
}

// agg[(et*N+dst)*128 + d] += x[src*128 + d]   (32 lanes per edge, 4 floats each)
__global__ __launch_bounds__(256) void scatter_kernel(
    const int* __restrict__ src, const int* __restrict__ dst,
    const int* __restrict__ et, const float* __restrict__ x,
    float* __restrict__ agg, int E, int n) {
  int gid = blockIdx.x * blockDim.x + threadIdx.x;
  int e = gid >> 5;
  if (e >= E) return;
  int part = (gid & 31) << 2;                 // 0,4,...,124
  int s = src[e], d0 = dst[e], r = et[e];
  const float4 v = *(const float4*)(x + (size_t)s * DIM + part);
  float* o = agg + ((size_t)r * n + d0) * DIM + part;
  atomic_fadd(o + 0, v.x);
  atomic_fadd(o + 1, v.y);
  atomic_fadd(o + 2, v.z);
  atomic_fadd(o + 3, v.w);
}

// out[n,:] = relu?( A[n,:] @ Bcat + bias )  where
//   A[n, 0:384]  = agg[r*N+n,:]*sc[r*N+n]  (r = seg/128)
//   A[n, 384:512]= xin[n,:]
//   Bcat rows 0:384 = W (flat [3*128,128]), rows 384:512 = root
// One wave computes 16 rows x 128 cols via 8 fp32 WMMA accumulators.
template <int RELU>
__global__ __launch_bounds__(256) void rgcn_gemm_kernel(
    const float* __restrict__ agg, const float* __restrict__ sc,
    const float* xin,
    const float* __restrict__ W, const float* __restrict__ root,
    const float* __restrict__ bias, float* out, int n) {
  // 64 rows x 128 cols staged per chunk, K-paired layout:
  // lds[(row>>1)*258 + col*2 + (row&1)]  (pair stride 258 staggers banks)
  __shared__ float ldsB[32 * 258];

  const int lane = threadIdx.x & 31;
  const int wave = threadIdx.x >> 5;
  const int lrow = lane & 15;
  const int hi = lane >> 4;  // 0: K+0/K+1 halves, 1: K+2/K+3 halves
  int rowA = blockIdx.x * 128 + wave * 16 + lrow;
  if (rowA >= n) rowA = n - 1;  // clamp loads; stores are guarded

  v8f acc[8] = {};

  for (int kc = 0; kc < KTOT; kc += 64) {
    __syncthreads();
    // Cooperative load of 64x128 weight rows into LDS (paired layout)
    for (int i = threadIdx.x; i < 64 * DIM; i += 256) {
      int r = i >> 7;
      int c = i & 127;
      int kg = kc + r;
      float v = (kg < 384) ? W[(size_t)kg * DIM + c]
                           : root[(size_t)(kg - 384) * DIM + c];
      ldsB[(r >> 1) * 258 + c * 2 + (r & 1)] = v;
    }

    // Chunk-uniform A source (chunks never straddle a relation boundary)
    int cc = kc >> 6;
    const float* aptr;
    float s = 1.0f;
    if (cc < 6) {
      int r = cc >> 1;
      aptr = agg + ((size_t)r * n + rowA) * DIM + (cc & 1) * 64;
      s = sc[r * n + rowA];
    } else {
      aptr = xin + (size_t)rowA * DIM + (cc - 6) * 64;
    }

    // Batch-prefetch all 16 A fragments for this chunk (one loadcnt wait),
    // issued while the LDS fill above is still in flight.
    v2f a[16];
#pragma unroll
    for (int ks = 0; ks < 16; ks++) {
      a[ks] = *(const v2f*)(aptr + ks * 4 + hi * 2);
    }
#pragma unroll
    for (int ks = 0; ks < 16; ks++) {
      a[ks] = a[ks] * s;
    }

    __syncthreads();  // LDS B chunk ready

#pragma unroll
    for (int ks = 0; ks < 16; ks++) {
      const int kl = ks * 4;
      const float* bp = &ldsB[((kl >> 1) + hi) * 258 + lrow * 2];
      // Gather all 8 B fragments first so DS loads pipeline ahead of WMMA.
      v2f b[8];
#pragma unroll
      for (int t = 0; t < 8; t++) {
        b[t] = *(const v2f*)(bp + t * 32);
      }
#pragma unroll
      for (int t = 0; t < 8; t++) {
        acc[t] = __builtin_amdgcn_wmma_f32_16x16x4_f32(
            false, a[ks], false, b[t], (short)0, acc[t], false, false);
      }
    }
  }

  // Epilogue: C/D layout — VGPR j: lanes 0-15 -> M=j, lanes 16-31 -> M=j+8
  int baseRow = blockIdx.x * 128 + wave * 16 + hi * 8;
#pragma unroll
  for (int t = 0; t < 8; t++) {
    int col = t * 16 + lrow;
    float bv = bias[col];
#pragma unroll
    for (int j = 0; j < 8; j++) {
      int rr = baseRow + j;
      if (rr < n) {
        float v = acc[t][j] + bv;
        if (RELU) v = fmaxf(v, 0.0f);
        out[(size_t)rr * DIM + col] = v;
      }
    }
  }
}

extern "C" void kernel_launch(void* const* d_in, const int* in_sizes, int n_in,
                              void* d_out, int out_size, void* d_ws, size_t ws_size,
                              hipStream_t stream) {
  const int* edge_index = (const int*)d_in[0];   // [2, E]
  const int* edge_type  = (const int*)d_in[1];   // [E]
  const int* roles      = (const int*)d_in[2];   // [N]
  const float* id_emb   = (const float*)d_in[3]; // [N, 112]
  const float* role_emb = (const float*)d_in[4]; // [4, 16]
  const float* W1    = (const float*)d_in[5];    // [3,128,128]
  const float* root1 = (const float*)d_in[6];    // [128,128]
  const float* b1    = (const float*)d_in[7];    // [128]
  const float* W2    = (const float*)d_in[8];
  const float* root2 = (const float*)d_in[9];
  const float* b2    = (const float*)d_in[10];
  float* out = (float*)d_out;

  const int E = in_sizes[1];
  const int n = in_sizes[2];
  const int* src = edge_index;
  const int* dst = edge_index + E;

  // Workspace: x [N*128] | agg [3N*128] | sc/cnt [3N]   (~206 MB)
  float* xbuf = (float*)d_ws;
  float* agg  = xbuf + (size_t)n * DIM;
  float* scb  = agg + (size_t)NREL * n * DIM;

  hipMemsetAsync(scb, 0, (size_t)NREL * n * sizeof(int), stream);
  hipMemsetAsync(agg, 0, (size_t)NREL * n * DIM * sizeof(float), stream);

  int tot = n * DIM;
  build_x_kernel<<<(tot + 255) / 256, 256, 0, stream>>>(id_emb, role_emb, roles, xbuf, n);
  count_kernel<<<(E + 255) / 256, 256, 0, stream>>>(dst, edge_type, (int*)scb, E, n);
  scale_kernel<<<(NREL * n + 255) / 256, 256, 0, stream>>>(scb, NREL * n);

  int sthreads = E * 32;
  int gblocks = (n + 127) / 128;

  // Layer 1: agg(x) -> GEMM -> relu -> h (stored in d_out)
  scatter_kernel<<<(sthreads + 255) / 256, 256, 0, stream>>>(src, dst, edge_type, xbuf, agg, E, n);
  rgcn_gemm_kernel<1><<<gblocks, 256, 0, stream>>>(agg, scb, xbuf, W1, root1, b1, out, n);

  // Layer 2: agg(h) -> GEMM -> d_out (in-place: x-part reads are row-local)
  hipMemsetAsync(agg, 0, (size_t)NREL * n * DIM * sizeof(float), stream);
  scatter_kernel<<<(sthreads + 255) / 256, 256, 0, stream>>>(src, dst, edge_type, out, agg, E, n);
  rgcn_gemm_kernel<0><<<gblocks, 256, 0, stream>>>(agg, scb, out, W2, root2, b2, out, n);
}